// AffinityFC_51118700757140
// MI455X (gfx1250) — compile-verified
//
#include <hip/hip_runtime.h>
#include <hip/hip_bf16.h>

typedef float v2f __attribute__((ext_vector_type(2)));
typedef float v4f __attribute__((ext_vector_type(4)));
typedef float v8f __attribute__((ext_vector_type(8)));

// ---------------------------------------------------------------------------
// Phase 1: px[b,n,h] = X·W1x^T + b1  and  py[b,m,h] = Y·W1y^T
// One wave32 computes a 16x32 f32 tile via V_WMMA_F32_16X16X4_F32 (two 16x16
// accumulators sharing the A fragment -> 1 A load + 2 B loads per 2 WMMAs).
// A (16x4 f32) lane layout: lane L holds M = L%16, K = 2*(L/16)+{0,1}  -> b64
// B (4x16 f32) lane layout: lane L holds N = L%16, K = 2*(L/16)+{0,1}  -> b64
// C/D (16x16 f32): VGPR i, lane L -> row i + 8*(L/16), col L%16
// ---------------------------------------------------------------------------
__global__ __launch_bounds__(32) void fc_wmma_f32(
    const float* __restrict__ X,   // (8*128, 512) flattened
    const float* __restrict__ Y,   // (8*128, 512) flattened
    const float* __restrict__ W1,  // (1024, 1024) row-major
    const float* __restrict__ b1,  // (1024)
    float* __restrict__ px,        // (8*128, 1024)
    float* __restrict__ py)        // (8*128, 1024)
{
    const int lane = threadIdx.x & 31;
    const int mt = blockIdx.x;        // row tile over B*N = 1024 rows (16 each)
    const int nt = blockIdx.y;        // col tile over hd = 1024 (32 each)
    const int z  = blockIdx.z;        // 0 -> px path, 1 -> py path

    const float* A = z ? Y : X;
    const int koff = z ? 512 : 0;     // W1y is columns [512,1024)
    float* D = z ? py : px;

    const int l16  = lane & 15;
    const int kp   = (lane >> 4) * 2;       // lane-half K offset
    const int rowA = mt * 16 + l16;         // A row (M)
    const int hh0  = nt * 32 + l16;         // first output column (N)
    const int hh1  = hh0 + 16;              // second output column

    const float* aptr  = A  + rowA * 512  + kp;
    const float* bptr0 = W1 + hh0 * 1024 + koff + kp;
    const float* bptr1 = W1 + hh1 * 1024 + koff + kp;

    v8f c0 = {0.f, 0.f, 0.f, 0.f, 0.f, 0.f, 0.f, 0.f};
    v8f c1 = {0.f, 0.f, 0.f, 0.f, 0.f, 0.f, 0.f, 0.f};
#pragma unroll 4
    for (int k = 0; k < 512; k += 4) {
        v2f a  = *(const v2f*)(aptr  + k);
        v2f b0 = *(const v2f*)(bptr0 + k);
        v2f b1v = *(const v2f*)(bptr1 + k);
        c0 = __builtin_amdgcn_wmma_f32_16x16x4_f32(
                 false, a, false, b0, (short)0, c0, false, false);
        c1 = __builtin_amdgcn_wmma_f32_16x16x4_f32(
                 false, a, false, b1v, (short)0, c1, false, false);
    }

    const float bias0 = z ? 0.0f : b1[hh0];
    const float bias1 = z ? 0.0f : b1[hh1];
    float* dptr = D + (mt * 16 + 8 * (lane >> 4)) * 1024 + nt * 32 + l16;
#pragma unroll
    for (int i = 0; i < 8; ++i) {
        dptr[i * 1024]      = c0[i] + bias0;
        dptr[i * 1024 + 16] = c1[i] + bias1;
    }
}

// ---------------------------------------------------------------------------
// Phase 2: out[b,n,m] = b2 + sum_h relu(px[b,n,h] + py[b,m,h]) * w2[h]
// Relu breaks bilinearity -> VALU reduction. 32x32 (n,m) tile per 256-thread
// block; h staged through LDS in chunks of 64.
//   py tile stored TRANSPOSED [h][m] (stride 36, 16B aligned) so a thread's
//     4 m-values at one h are a single ds_load_b128 (broadcast across lanes
//     that share cb -> no conflicts).
//   px tile row-major [n][h] (stride 65) so consecutive-h reads merge into
//     b64/b128 under unroll; w2s chunk likewise contiguous.
// Each thread owns 1 n x 4 m -> per 4 h: ~6 LDS ops vs 48 VALU ops.
// ---------------------------------------------------------------------------
__global__ __launch_bounds__(256) void pair_relu_reduce(
    const float* __restrict__ px,   // (8*128, 1024), b1 already folded in
    const float* __restrict__ py,   // (8*128, 1024)
    const float* __restrict__ W2,   // (1024)
    const float* __restrict__ b2,   // (1)
    float* __restrict__ out)        // (8, 128, 128)
{
    __shared__ float pxs[32 * 65];   // [n][h] padded
    __shared__ float pys[64 * 36];   // [h][m] transposed, padded stride 36
    __shared__ float w2s[64];

    const int tid = threadIdx.x;
    const int b   = blockIdx.z;
    const int n0  = blockIdx.x * 32;
    const int m0  = blockIdx.y * 32;
    const int r   = tid >> 3;          // n within tile: 0..31
    const int cb  = (tid & 7) * 4;     // m base within tile: 0,4,...,28

    const float* pxb = px + (b * 128 + n0) * 1024;
    const float* pyb = py + (b * 128 + m0) * 1024;

    float acc0 = 0.f, acc1 = 0.f, acc2 = 0.f, acc3 = 0.f;

    for (int h0 = 0; h0 < 1024; h0 += 64) {
#pragma unroll
        for (int idx = tid; idx < 2048; idx += 256) {
            const int row = idx >> 6, h = idx & 63;
            pxs[row * 65 + h]  = pxb[row * 1024 + h0 + h];
            pys[h * 36 + row]  = pyb[row * 1024 + h0 + h];  // transpose
        }
        if (tid < 64) w2s[tid] = W2[h0 + tid];
        __syncthreads();

#pragma unroll 4
        for (int h = 0; h < 64; ++h) {
            const v4f  py4 = *(const v4f*)&pys[h * 36 + cb];  // ds_load_b128
            const float pxv = pxs[r * 65 + h];
            const float w   = w2s[h];
            acc0 = fmaf(fmaxf(pxv + py4[0], 0.f), w, acc0);
            acc1 = fmaf(fmaxf(pxv + py4[1], 0.f), w, acc1);
            acc2 = fmaf(fmaxf(pxv + py4[2], 0.f), w, acc2);
            acc3 = fmaf(fmaxf(pxv + py4[3], 0.f), w, acc3);
        }
        __syncthreads();
    }

    const float bb = b2[0];
    float* o = out + b * (128 * 128) + (n0 + r) * 128 + m0 + cb;
    o[0] = acc0 + bb;
    o[1] = acc1 + bb;
    o[2] = acc2 + bb;
    o[3] = acc3 + bb;
}

extern "C" void kernel_launch(void* const* d_in, const int* in_sizes, int n_in,
                              void* d_out, int out_size, void* d_ws, size_t ws_size,
                              hipStream_t stream) {
    (void)in_sizes; (void)n_in; (void)out_size; (void)ws_size;

    const float* X  = (const float*)d_in[0];   // (8,128,512)
    const float* Y  = (const float*)d_in[1];   // (8,128,512)
    const float* W1 = (const float*)d_in[2];   // (1024,1024)
    const float* b1 = (const float*)d_in[3];   // (1024)
    const float* W2 = (const float*)d_in[4];   // (1,1024)
    const float* b2 = (const float*)d_in[5];   // (1)
    float* out = (float*)d_out;                // (8,128,128)

    float* px = (float*)d_ws;                  // 8*128*1024 f32 = 4 MB
    float* py = px + 8 * 128 * 1024;           // 4 MB

    // Phase 1: 64 row-tiles x 32 col-tiles (16x32 per wave), z: px vs py.
    dim3 g1(64, 32, 2);
    fc_wmma_f32<<<g1, 32, 0, stream>>>(X, Y, W1, b1, px, py);

    // Phase 2: 32x32 (n,m) tiles per block, one block-z per batch.
    dim3 g2(4, 4, 8);
    pair_relu_reduce<<<g2, 256, 0, stream>>>(px, py, W2, b2, out);
}